// PoseUpdate_78958678770056
// MI455X (gfx1250) — compile-verified
//
#include <hip/hip_runtime.h>
#include <hip/hip_bf16.h>

typedef float v2f __attribute__((ext_vector_type(2)));
typedef float v8f __attribute__((ext_vector_type(8)));

#define Bn   16
#define Cn   16
#define Hn   96
#define Wn   320
#define HWn  (Hn * Wn)      // 30720
#define TPB  256            // 8 wave32
#define PPT  4              // pixels per thread
#define NBLK 30             // NBLK*TPB*PPT == HWn

// ---------------------------------------------------------------------------
// Kernel 1: per-pixel Gauss-Newton accumulation.
// Per-lane: 28-scalar upper-triangular accumulator of the augmented 7x7
//   [ H  b ; b' rr ]  system. Channel loop reduced to 6 scalars via
//   H_p = w * Jpix' * (Sum_c v v') * Jpix  factorization.
// Cross-lane reduction: V_WMMA_F32_16X16X4_F32 with A = ones, B rows =
//   per-lane accumulator vectors staged in LDS (matrix-pipe tree-sum).
// ---------------------------------------------------------------------------
__global__ __launch_bounds__(TPB) void pose_hb_kernel(
    const float* __restrict__ p2,
    const float* __restrict__ P2,
    const float* __restrict__ weight,
    const float* __restrict__ src_feat,
    const float* __restrict__ tgt_feat,
    const float* __restrict__ src_w,
    const float* __restrict__ tgt_w,
    const float* __restrict__ calibK,
    float* __restrict__ partials)       // [Bn][NBLK][28]
{
    __shared__ float red[TPB][33];      // 33: bank-conflict pad
    __shared__ float wred[TPB / 32][32];

    const int b    = blockIdx.y;
    const int tid  = threadIdx.x;
    const int lane = tid & 31;
    const int wave = tid >> 5;

    const float fx = calibK[b * 9 + 0];
    const float fy = calibK[b * 9 + 4];

    // p2 layout: (B, 2, 1, 5, H, W); we only need the s=0 (center) plane.
    const float* p2x = p2 + (size_t)b * 10 * HWn;
    const float* p2y = p2 + ((size_t)b * 10 + 5) * HWn;
    const float* P2b = P2 + (size_t)b * 3 * HWn;
    const float* wgb = weight   + (size_t)b * HWn;
    const float* swb = src_w    + (size_t)b * HWn;
    const float* twb = tgt_w    + (size_t)b * HWn;
    const float* sfb = src_feat + (size_t)b * Cn * HWn;
    const float* tfb = tgt_feat + (size_t)b * Cn * HWn;

    float acc[28];
#pragma unroll
    for (int i = 0; i < 28; ++i) acc[i] = 0.0f;

#pragma unroll 1
    for (int it = 0; it < PPT; ++it) {
        const int p = (blockIdx.x * PPT + it) * TPB + tid;   // coalesced

        const float x   = p2x[p];
        const float y   = p2y[p];
        const float x0f = floorf(x);
        const float y0f = floorf(y);
        const float wx  = x - x0f;
        const float wy  = y - y0f;
        int x0 = (int)x0f;
        int y0 = (int)y0f;
        // inputs are interior (xc in [2,W-3]); clamp is a free OOB guard
        x0 = min(max(x0, 1), Wn - 3);
        y0 = min(max(y0, 1), Hn - 3);
        const float u0 = 1.0f - wx, u1 = wx;
        const float v0 = 1.0f - wy, v1 = wy;

        // bilinear sample of tgt_w at the center point
        const float* tw0 = twb + y0 * Wn + x0;
        const float  twv = v0 * (u0 * tw0[0]  + u1 * tw0[1]) +
                           v1 * (u0 * tw0[Wn] + u1 * tw0[Wn + 1]);
        const float  wgt = swb[p] * twv * wgb[p];

        // channel loop: one 4x4 patch per channel covers all 5 sample points
        float Sxx = 0, Sxy = 0, Syy = 0, Srx = 0, Sry = 0, Srr = 0;
        const float* tc = tfb + (y0 - 1) * Wn + (x0 - 1);
#pragma unroll 4
        for (int c = 0; c < Cn; ++c) {
            const float* t0r = tc + (size_t)c * HWn;
            const float* t1r = t0r + Wn;
            const float* t2r = t1r + Wn;
            const float* t3r = t2r + Wn;
            const float t01 = t0r[1], t02 = t0r[2];
            const float t10 = t1r[0], t11 = t1r[1], t12 = t1r[2], t13 = t1r[3];
            const float t20 = t2r[0], t21 = t2r[1], t22 = t2r[2], t23 = t2r[3];
            const float t31 = t3r[1], t32 = t3r[2];

            const float r0 = u0 * t01 + u1 * t02;
            const float r1 = u0 * t11 + u1 * t12;
            const float r2 = u0 * t21 + u1 * t22;
            const float r3 = u0 * t31 + u1 * t32;
            const float f0 = v0 * r1 + v1 * r2;                     // center
            const float d1 = u0 * (t12 - t10) + u1 * (t13 - t11);
            const float d2 = u0 * (t22 - t20) + u1 * (t23 - t21);
            const float gx = 0.5f * (v0 * d1 + v1 * d2);            // (f1-f2)/2
            const float gy = 0.5f * (v0 * (r2 - r0) + v1 * (r3 - r1)); // (f3-f4)/2
            const float rs = sfb[(size_t)c * HWn + p] - f0;         // residual

            Sxx = fmaf(gx, gx, Sxx);
            Sxy = fmaf(gx, gy, Sxy);
            Syy = fmaf(gy, gy, Syy);
            Srx = fmaf(rs, gx, Srx);
            Sry = fmaf(rs, gy, Sry);
            Srr = fmaf(rs, rs, Srr);
        }

        // projection Jacobian rows (2x6)
        const float X  = P2b[p];
        const float Y  = P2b[HWn + p];
        const float Z  = P2b[2 * HWn + p];
        const float iZ = 1.0f / Z;
        const float fxZ = fx * iZ, fyZ = fy * iZ;
        const float fxXZ2 = fxZ * X * iZ;
        const float fyYZ2 = fyZ * Y * iZ;

        const float Pv[6] = { fxZ, 0.0f, -fxXZ2, -fxXZ2 * Y, fx + fxXZ2 * X, -fxZ * Y };
        const float Qv[6] = { 0.0f, fyZ, -fyYZ2, -fy - fyYZ2 * Y, fyYZ2 * X, fyZ * X };

        // rank-2 update of augmented 7x7 (upper triangle, b as column 6)
        int o = 0;
#pragma unroll
        for (int k = 0; k < 6; ++k) {
            const float Ak = wgt * (Sxx * Pv[k] + Sxy * Qv[k]);
            const float Bk = wgt * (Sxy * Pv[k] + Syy * Qv[k]);
#pragma unroll
            for (int l = k; l < 6; ++l) {
                acc[o] = fmaf(Ak, Pv[l], fmaf(Bk, Qv[l], acc[o]));
                ++o;
            }
            acc[o] = fmaf(wgt, Srx * Pv[k] + Sry * Qv[k], acc[o]);  // b entry
            ++o;
        }
        acc[27] = fmaf(wgt, Srr, acc[27]);
    }

    // stage per-lane accumulators (zero-pad cols 28..31)
#pragma unroll
    for (int i = 0; i < 28; ++i) red[tid][i] = acc[i];
    red[tid][28] = 0.0f; red[tid][29] = 0.0f;
    red[tid][30] = 0.0f; red[tid][31] = 0.0f;
    __syncthreads();

    // matrix-pipe cross-lane sum: C[m][n] = sum_k 1 * B[k][n]
    v8f c0 = {0, 0, 0, 0, 0, 0, 0, 0};
    v8f c1 = {0, 0, 0, 0, 0, 0, 0, 0};
    const v2f ones = {1.0f, 1.0f};
    const int half    = lane >> 4;   // A/B VGPR0 holds K=0 (lanes 0-15), K=2 (16-31)
    const int col     = lane & 15;
    const int rowbase = wave * 32;
#pragma unroll
    for (int g = 0; g < 8; ++g) {
        const int r = rowbase + g * 4 + half * 2;
        v2f b0, b1;
        b0.x = red[r][col];          b0.y = red[r + 1][col];
        b1.x = red[r][16 + col];     b1.y = red[r + 1][16 + col];
        c0 = __builtin_amdgcn_wmma_f32_16x16x4_f32(false, ones, false, b0,
                                                   (short)0, c0, false, false);
        c1 = __builtin_amdgcn_wmma_f32_16x16x4_f32(false, ones, false, b1,
                                                   (short)0, c1, false, false);
    }

    // C row 0 (VGPR 0, lanes 0-15) holds the wave-wide column sums
    if (lane < 16) {
        wred[wave][lane]      = c0[0];
        wred[wave][16 + lane] = c1[0];
    }
    __syncthreads();

    if (tid < 28) {
        float s = 0.0f;
#pragma unroll
        for (int w2 = 0; w2 < TPB / 32; ++w2) s += wred[w2][tid];
        partials[((size_t)b * NBLK + blockIdx.x) * 28 + tid] = s;  // no atomics
    }
}

// ---------------------------------------------------------------------------
// Kernel 2: sum partials, pivoted 6x6 solve, se3_exp, pose composition.
// ---------------------------------------------------------------------------
__device__ __forceinline__ int triidx(int k, int l) {
    if (k > l) { int t = k; k = l; l = t; }
    return k * 7 - (k * (k + 1)) / 2 + l;
}

__global__ void pose_solve_kernel(const float* __restrict__ partials,
                                  const float* __restrict__ poses,
                                  float* __restrict__ out)
{
    const int b = threadIdx.x;
    if (b >= Bn) return;

    float S[28];
#pragma unroll
    for (int i = 0; i < 28; ++i) S[i] = 0.0f;
    for (int j = 0; j < NBLK; ++j) {
        const float* pp = partials + ((size_t)b * NBLK + j) * 28;
#pragma unroll
        for (int i = 0; i < 28; ++i) S[i] += pp[i];
    }

    float M[6][7];
    for (int k = 0; k < 6; ++k)
        for (int l = 0; l < 7; ++l)
            M[k][l] = S[triidx(k, l)];     // col 6 is b

    // Gaussian elimination with partial pivoting
    for (int cI = 0; cI < 6; ++cI) {
        int piv = cI; float mx = fabsf(M[cI][cI]);
        for (int r = cI + 1; r < 6; ++r) {
            float v = fabsf(M[r][cI]);
            if (v > mx) { mx = v; piv = r; }
        }
        if (piv != cI)
            for (int l = cI; l < 7; ++l) {
                float t = M[cI][l]; M[cI][l] = M[piv][l]; M[piv][l] = t;
            }
        const float inv = 1.0f / M[cI][cI];
        for (int r = cI + 1; r < 6; ++r) {
            const float f = M[r][cI] * inv;
            for (int l = cI; l < 7; ++l) M[r][l] -= f * M[cI][l];
        }
    }
    float upd[6];
    for (int r = 5; r >= 0; --r) {
        float s = M[r][6];
        for (int l = r + 1; l < 6; ++l) s -= M[r][l] * upd[l];
        upd[r] = s / M[r][r];
    }

    // se3_exp(upd)
    const float rx = upd[0], ry = upd[1], rz = upd[2];
    const float pa = upd[3], pb = upd[4], pc = upd[5];
    const float th2   = pa * pa + pb * pb + pc * pc;
    const bool  small = th2 < 1e-8f;
    const float th2s  = small ? 1.0f : th2;
    const float th    = sqrtf(th2s);
    const float Ae = small ? 1.0f - th2 * (1.0f / 6.0f)   : sinf(th) / th;
    const float Be = small ? 0.5f - th2 * (1.0f / 24.0f)  : (1.0f - cosf(th)) / th2s;
    const float Ce = small ? (1.0f / 6.0f) - th2 * (1.0f / 120.0f)
                           : (th - sinf(th)) / (th2s * th);

    const float K[3][3] = {{0, -pc, pb}, {pc, 0, -pa}, {-pb, pa, 0}};
    float K2[3][3];
    for (int i = 0; i < 3; ++i)
        for (int j = 0; j < 3; ++j)
            K2[i][j] = K[i][0] * K[0][j] + K[i][1] * K[1][j] + K[i][2] * K[2][j];

    float R[3][3], V[3][3];
    for (int i = 0; i < 3; ++i)
        for (int j = 0; j < 3; ++j) {
            const float id = (i == j) ? 1.0f : 0.0f;
            R[i][j] = id + Ae * K[i][j] + Be * K2[i][j];
            V[i][j] = id + Be * K[i][j] + Ce * K2[i][j];
        }
    float t[3];
    for (int i = 0; i < 3; ++i)
        t[i] = V[i][0] * rx + V[i][1] * ry + V[i][2] * rz;

    // new_pose = T_exp @ poses[b]
    const float* P = poses + b * 16;
    float O[16];
    for (int i = 0; i < 3; ++i)
        for (int j = 0; j < 4; ++j)
            O[i * 4 + j] = R[i][0] * P[0 + j] + R[i][1] * P[4 + j] +
                           R[i][2] * P[8 + j] + t[i] * P[12 + j];
    for (int j = 0; j < 4; ++j) O[12 + j] = P[12 + j];

    for (int i = 0; i < 16; ++i) out[b * 16 + i] = O[i];
    for (int i = 0; i < 6;  ++i) out[Bn * 16 + b * 6 + i] = upd[i];
}

// ---------------------------------------------------------------------------
extern "C" void kernel_launch(void* const* d_in, const int* in_sizes, int n_in,
                              void* d_out, int out_size, void* d_ws, size_t ws_size,
                              hipStream_t stream)
{
    const float* poses    = (const float*)d_in[0];
    const float* calib_K  = (const float*)d_in[1];
    const float* p2       = (const float*)d_in[2];
    const float* P2       = (const float*)d_in[3];
    const float* weight   = (const float*)d_in[4];
    const float* src_feat = (const float*)d_in[5];
    const float* tgt_feat = (const float*)d_in[6];
    const float* src_w    = (const float*)d_in[7];
    const float* tgt_w    = (const float*)d_in[8];
    float* out      = (float*)d_out;
    float* partials = (float*)d_ws;      // Bn*NBLK*28 floats (53760 bytes)

    dim3 grid(NBLK, Bn);
    pose_hb_kernel<<<grid, TPB, 0, stream>>>(p2, P2, weight, src_feat, tgt_feat,
                                             src_w, tgt_w, calib_K, partials);
    pose_solve_kernel<<<1, 32, 0, stream>>>(partials, poses, out);
}